// PointNetSetAbstraction_72524817760258
// MI455X (gfx1250) — compile-verified
//
#include <hip/hip_runtime.h>

typedef float v2f __attribute__((ext_vector_type(2)));
typedef float v8f __attribute__((ext_vector_type(8)));

#define BATCH   16
#define NPTS    4096
#define NPOINT  512
#define NSAMPLE 32
#define MROWS   (BATCH*NPOINT*NSAMPLE)   // 262144
#define RAD2    0.25f
#define NCHUNK  (MROWS/128)              // 2048 GEMM blocks = BN partial chunks

// ---------------------------------------------------------------------------
// Farthest point sampling: one block per batch, points resident in LDS,
// per-thread running distances in registers, LDS tree argmax per step.
// ---------------------------------------------------------------------------
__global__ __launch_bounds__(1024)
void fps_kernel(const float* __restrict__ xyz, int* __restrict__ fps_idx) {
    __shared__ float px[NPTS], py[NPTS], pz[NPTS];
    __shared__ float rv[1024];
    __shared__ int   ri[1024];
    __shared__ int   far_s;
    const int b = blockIdx.x;
    const int t = threadIdx.x;
    const float* xb = xyz + (size_t)b * 3 * NPTS;
    for (int n = t; n < NPTS; n += 1024) {
        px[n] = xb[n];
        py[n] = xb[NPTS + n];
        pz[n] = xb[2 * NPTS + n];
    }
    float dist[4];
#pragma unroll
    for (int j = 0; j < 4; ++j) dist[j] = 1e10f;
    if (t == 0) far_s = 0;
    __syncthreads();

    for (int step = 0; step < NPOINT; ++step) {
        const int far = far_s;
        if (t == 0) fps_idx[b * NPOINT + step] = far;   // emit BEFORE update (scan semantics)
        const float cx = px[far], cy = py[far], cz = pz[far];
        float bv = -1.0f; int bi = 0;
#pragma unroll
        for (int j = 0; j < 4; ++j) {
            const int n = t + j * 1024;
            const float dx = px[n] - cx, dy = py[n] - cy, dz = pz[n] - cz;
            const float d  = dx * dx + dy * dy + dz * dz;
            const float dn = fminf(dist[j], d);
            dist[j] = dn;
            if (dn > bv) { bv = dn; bi = n; }            // ascending n -> first max kept
        }
        rv[t] = bv; ri[t] = bi;
        __syncthreads();
        for (int s = 512; s > 0; s >>= 1) {
            if (t < s) {
                const float v2 = rv[t + s]; const int i2 = ri[t + s];
                if (v2 > rv[t] || (v2 == rv[t] && i2 < ri[t])) { rv[t] = v2; ri[t] = i2; }
            }
            __syncthreads();
        }
        if (t == 0) far_s = ri[0];
        __syncthreads();
    }
}

// ---------------------------------------------------------------------------
// Ball query + grouping: one block per batch, one thread per query point.
// Scans indices ascending, keeps first NSAMPLE within radius, pads with the
// first neighbor (reference sort/pad semantics). Writes x0 with K padded
// 9 -> 12 (zeros) and new_xyz (B,3,512).
// ---------------------------------------------------------------------------
__device__ __forceinline__
void write_row(float* __restrict__ x0, size_t row, const float* __restrict__ fb,
               int n, float dx, float dy, float dz) {
    float* r = x0 + row * 12;
    r[0] = dx; r[1] = dy; r[2] = dz;
    r[3] = fb[n];             r[4] = fb[NPTS + n];     r[5] = fb[2 * NPTS + n];
    r[6] = fb[3 * NPTS + n];  r[7] = fb[4 * NPTS + n]; r[8] = fb[5 * NPTS + n];
    r[9] = 0.f; r[10] = 0.f; r[11] = 0.f;
}

__global__ __launch_bounds__(512)
void group_kernel(const float* __restrict__ xyz, const float* __restrict__ feat,
                  const int* __restrict__ fps_idx, float* __restrict__ x0,
                  float* __restrict__ out_xyz) {
    __shared__ float px[NPTS], py[NPTS], pz[NPTS];
    const int b = blockIdx.x;
    const int t = threadIdx.x;                 // query index k
    const float* xb = xyz + (size_t)b * 3 * NPTS;
    for (int n = t; n < NPTS; n += 512) {
        px[n] = xb[n]; py[n] = xb[NPTS + n]; pz[n] = xb[2 * NPTS + n];
    }
    __syncthreads();
    const int fi = fps_idx[b * NPOINT + t];
    const float qx = px[fi], qy = py[fi], qz = pz[fi];
    out_xyz[(b * 3 + 0) * NPOINT + t] = qx;
    out_xyz[(b * 3 + 1) * NPOINT + t] = qy;
    out_xyz[(b * 3 + 2) * NPOINT + t] = qz;

    const float* fb = feat + (size_t)b * 6 * NPTS;
    const size_t rowbase = ((size_t)(b * NPOINT + t)) * NSAMPLE;
    int cnt = 0; int nfirst = 0;
    for (int n = 0; n < NPTS && cnt < NSAMPLE; ++n) {
        const float dx = px[n] - qx, dy = py[n] - qy, dz = pz[n] - qz;
        const float d = dx * dx + dy * dy + dz * dz;
        if (d <= RAD2) {
            if (cnt == 0) nfirst = n;
            write_row(x0, rowbase + cnt, fb, n, dx, dy, dz);
            ++cnt;
        }
    }
    const float fdx = px[nfirst] - qx, fdy = py[nfirst] - qy, fdz = pz[nfirst] - qz;
    for (int j = cnt; j < NSAMPLE; ++j)
        write_row(x0, rowbase + j, fb, nfirst, fdx, fdy, fdz);
}

// ---------------------------------------------------------------------------
// Weight prep: W (cout x cin) -> Wt (ks x cout), k-major, zero-padded.
// ---------------------------------------------------------------------------
__global__ void wprep_kernel(const float* __restrict__ W, float* __restrict__ Wt,
                             int cin, int ks, int cout) {
    const int i = blockIdx.x * 256 + threadIdx.x;
    if (i < ks * cout) {
        const int c = i / cout, o = i % cout;
        Wt[i] = (c < cin) ? W[o * cin + c] : 0.f;
    }
}

// ---------------------------------------------------------------------------
// Fused GEMM: Y[m,o] = sum_c relu_bn(X)[m,c]*Wt[c,o] + bias[o]
//   via v_wmma_f32_16x16x4_f32, plus deterministic BN partial sums of Y.
// Block = 256 threads (8 waves) owns 128 rows; X tile staged in LDS with
// coalesced float4 loads (optionally applying previous-layer BN+ReLU).
// Each wave computes 16 rows x COUT, reusing the A fragment across all
// column tiles. Layouts per CDNA5 ISA 7.12.2:
//   A 16x4 : lane l, vgpr v -> m=(l&15), k=(l>>4)*2+v
//   B 4x16 : lane l, vgpr v -> n=(l&15), k=(l>>4)*2+v
//   D 16x16: lane l, vgpr v -> n=(l&15), m=v+(l>>4)*8
// ---------------------------------------------------------------------------
template<int KS, int COUT>
__global__ __launch_bounds__(256)
void gemm_fused_kernel(const float* __restrict__ X, const float* __restrict__ Wt,
                       const float* __restrict__ bias,
                       const float* __restrict__ inscl, const float* __restrict__ insft,
                       float* __restrict__ Y,
                       float* __restrict__ psum, float* __restrict__ psumsq) {
    constexpr int XS = KS + 4;       // padded LDS stride: conflict-free b64 reads
    constexpr int OT = COUT / 16;
    __shared__ float xs[128 * XS];
    __shared__ float rsum[16 * COUT];
    __shared__ float rsq [16 * COUT];

    const int t    = threadIdx.x;
    const int lane = t & 31;
    const int wave = t >> 5;
    const int half = lane >> 4;
    const int l15  = lane & 15;
    const size_t row0 = (size_t)blockIdx.x * 128;

    // ---- stage X tile (fusing previous layer's BN + ReLU if given) --------
    const float4* Xg = (const float4*)(X + row0 * KS);
    constexpr int NV = 128 * KS / 4;
    for (int i = t; i < NV; i += 256) {
        float4 v = Xg[i];
        const int flat = i * 4;
        const int r = flat / KS;
        const int c = flat % KS;     // float4 never crosses rows (KS % 4 == 0)
        if (inscl) {
            v.x = fmaxf(v.x * inscl[c]     + insft[c],     0.f);
            v.y = fmaxf(v.y * inscl[c + 1] + insft[c + 1], 0.f);
            v.z = fmaxf(v.z * inscl[c + 2] + insft[c + 2], 0.f);
            v.w = fmaxf(v.w * inscl[c + 3] + insft[c + 3], 0.f);
        }
        float* d = &xs[r * XS + c];
        d[0] = v.x; d[1] = v.y; d[2] = v.z; d[3] = v.w;
    }
    __syncthreads();

    // ---- WMMA main loop ---------------------------------------------------
    v8f acc[OT];
#pragma unroll
    for (int ot = 0; ot < OT; ++ot) acc[ot] = (v8f){0.f,0.f,0.f,0.f,0.f,0.f,0.f,0.f};

    const float* arow = &xs[(wave * 16 + l15) * XS];
#pragma unroll
    for (int kk = 0; kk < KS; kk += 4) {
        const int ka = kk + half * 2;
        v2f a;
        a.x = arow[ka];
        a.y = arow[ka + 1];
#pragma unroll
        for (int ot = 0; ot < OT; ++ot) {
            const int o = ot * 16 + l15;
            v2f bf;
            bf.x = Wt[(size_t)ka * COUT + o];
            bf.y = Wt[(size_t)(ka + 1) * COUT + o];
            acc[ot] = __builtin_amdgcn_wmma_f32_16x16x4_f32(
                false, a, false, bf, (short)0, acc[ot], false, false);
        }
    }

    // ---- epilogue: bias, store Y, deterministic BN partials ---------------
    const size_t m0 = row0 + wave * 16 + half * 8;
#pragma unroll
    for (int ot = 0; ot < OT; ++ot) {
        const int o = ot * 16 + l15;
        const float bv = bias[o];
        float ls = 0.f, lq = 0.f;
#pragma unroll
        for (int v = 0; v < 8; ++v) {
            const float y = acc[ot][v] + bv;
            Y[(m0 + v) * COUT + o] = y;
            ls += y; lq += y * y;
        }
        rsum[(wave * 2 + half) * COUT + o] = ls;
        rsq [(wave * 2 + half) * COUT + o] = lq;
    }
    __syncthreads();
    if (t < COUT) {
        float s = 0.f, q = 0.f;
#pragma unroll
        for (int i = 0; i < 16; ++i) { s += rsum[i * COUT + t]; q += rsq[i * COUT + t]; }
        psum  [(size_t)blockIdx.x * COUT + t] = s;
        psumsq[(size_t)blockIdx.x * COUT + t] = q;
    }
}

// ---------------------------------------------------------------------------
// BN stats finalize (deterministic fixed-order reduction over chunks).
// ---------------------------------------------------------------------------
__global__ void bnstats_kernel(const float* __restrict__ psum, const float* __restrict__ psumsq,
                               const float* __restrict__ g, const float* __restrict__ beta,
                               float* __restrict__ scale, float* __restrict__ shift,
                               int cout, int nchunk) {
    const int o = threadIdx.x;
    if (o >= cout) return;
    float s = 0.f, s2 = 0.f;
    for (int p = 0; p < nchunk; ++p) { s += psum[(size_t)p * cout + o]; s2 += psumsq[(size_t)p * cout + o]; }
    const float inv_n = 1.0f / (float)MROWS;
    const float mu  = s * inv_n;
    const float var = s2 * inv_n - mu * mu;
    const float r   = rsqrtf(var + 1e-5f);
    const float a   = g[o] * r;
    scale[o] = a;
    shift[o] = beta[o] - mu * a;
}

// ---------------------------------------------------------------------------
// Last layer: BN + ReLU + max over 32 samples + transpose to (B,128,NPOINT)
// ---------------------------------------------------------------------------
__global__ __launch_bounds__(256)
void final_kernel(const float* __restrict__ Y2, const float* __restrict__ scale,
                  const float* __restrict__ shift, float* __restrict__ out_feat) {
    const int i = blockIdx.x * 256 + threadIdx.x;   // B*NPOINT*128 threads
    const int o   = i & 127;
    const int kpt = (i >> 7) & (NPOINT - 1);
    const int b   = i >> 16;
    const float* base = Y2 + ((size_t)((b * NPOINT + kpt) * NSAMPLE)) * 128 + o;
    const float a = scale[o], c = shift[o];
    float m = 0.f;                                  // ReLU outputs are >= 0
#pragma unroll 4
    for (int s = 0; s < NSAMPLE; ++s)
        m = fmaxf(m, fmaxf(base[(size_t)s * 128] * a + c, 0.f));
    out_feat[((size_t)(b * 128 + o)) * NPOINT + kpt] = m;
}

// ---------------------------------------------------------------------------
extern "C" void kernel_launch(void* const* d_in, const int* in_sizes, int n_in,
                              void* d_out, int out_size, void* d_ws, size_t ws_size,
                              hipStream_t stream) {
    (void)in_sizes; (void)n_in; (void)out_size; (void)ws_size;
    const float* xyz  = (const float*)d_in[0];
    const float* feat = (const float*)d_in[1];
    const float* W0 = (const float*)d_in[2];
    const float* b0 = (const float*)d_in[3];
    const float* g0 = (const float*)d_in[4];
    const float* be0= (const float*)d_in[5];
    const float* W1 = (const float*)d_in[6];
    const float* b1 = (const float*)d_in[7];
    const float* g1 = (const float*)d_in[8];
    const float* be1= (const float*)d_in[9];
    const float* W2 = (const float*)d_in[10];
    const float* b2 = (const float*)d_in[11];
    const float* g2 = (const float*)d_in[12];
    const float* be2= (const float*)d_in[13];

    float* out      = (float*)d_out;
    float* out_xyz  = out;                       // (16,3,512)
    float* out_feat = out + BATCH * 3 * NPOINT;  // (16,128,512)

    char* ws = (char*)d_ws;
    size_t off = 0;
    auto alloc = [&](size_t bytes) -> void* {
        void* p = ws + off;
        off = (off + bytes + 255) & ~(size_t)255;
        return p;
    };
    int*   fpsI   = (int*)  alloc((size_t)BATCH * NPOINT * sizeof(int));
    float* Wt0    = (float*)alloc(12 * 64  * sizeof(float));
    float* Wt1    = (float*)alloc(64 * 64  * sizeof(float));
    float* Wt2    = (float*)alloc(64 * 128 * sizeof(float));
    float* scl0   = (float*)alloc(64  * sizeof(float));
    float* sft0   = (float*)alloc(64  * sizeof(float));
    float* scl1   = (float*)alloc(64  * sizeof(float));
    float* sft1   = (float*)alloc(64  * sizeof(float));
    float* scl2   = (float*)alloc(128 * sizeof(float));
    float* sft2   = (float*)alloc(128 * sizeof(float));
    float* psum   = (float*)alloc((size_t)NCHUNK * 128 * sizeof(float));
    float* psumsq = (float*)alloc((size_t)NCHUNK * 128 * sizeof(float));
    float* X0 = (float*)alloc((size_t)MROWS * 12  * sizeof(float));
    float* Y0 = (float*)alloc((size_t)MROWS * 64  * sizeof(float));
    float* Y1 = (float*)alloc((size_t)MROWS * 64  * sizeof(float));
    float* Y2 = (float*)alloc((size_t)MROWS * 128 * sizeof(float));

    // --- sampling + grouping + weight prep ---------------------------------
    fps_kernel  <<<BATCH, 1024, 0, stream>>>(xyz, fpsI);
    wprep_kernel<<<(12 * 64  + 255) / 256, 256, 0, stream>>>(W0, Wt0, 9,  12, 64);
    wprep_kernel<<<(64 * 64  + 255) / 256, 256, 0, stream>>>(W1, Wt1, 64, 64, 64);
    wprep_kernel<<<(64 * 128 + 255) / 256, 256, 0, stream>>>(W2, Wt2, 64, 64, 128);
    group_kernel<<<BATCH, 512, 0, stream>>>(xyz, feat, fpsI, X0, out_xyz);

    // --- layer 0: (M x 12) x (12 x 64) -------------------------------------
    gemm_fused_kernel<12, 64><<<NCHUNK, 256, 0, stream>>>(
        X0, Wt0, b0, nullptr, nullptr, Y0, psum, psumsq);
    bnstats_kernel<<<1, 128, 0, stream>>>(psum, psumsq, g0, be0, scl0, sft0, 64, NCHUNK);

    // --- layer 1: relu_bn0(Y0) x (64 x 64) ---------------------------------
    gemm_fused_kernel<64, 64><<<NCHUNK, 256, 0, stream>>>(
        Y0, Wt1, b1, scl0, sft0, Y1, psum, psumsq);
    bnstats_kernel<<<1, 128, 0, stream>>>(psum, psumsq, g1, be1, scl1, sft1, 64, NCHUNK);

    // --- layer 2: relu_bn1(Y1) x (64 x 128) --------------------------------
    gemm_fused_kernel<64, 128><<<NCHUNK, 256, 0, stream>>>(
        Y1, Wt2, b2, scl1, sft1, Y2, psum, psumsq);
    bnstats_kernel<<<1, 128, 0, stream>>>(psum, psumsq, g2, be2, scl2, sft2, 128, NCHUNK);

    // --- fused BN + ReLU + maxpool + transpose -----------------------------
    final_kernel<<<(BATCH * NPOINT * 128) / 256, 256, 0, stream>>>(Y2, scl2, sft2, out_feat);
}